// ConvQuadInterp3d_89283780149979
// MI455X (gfx1250) — compile-verified
//
#include <hip/hip_runtime.h>

// ConvQuadInterp3d for x:(B=2,C=1,D=8,H=512,W=512) f32.
// Outputs concatenated: coords (B,C,3,D,H,W) then y_max (B,C,D,H,W).
//
// Memory-bound (~84 MB @ 23.3 TB/s => ~3.6us floor). CDNA5 paths used:
//   * global_load_async_to_lds_b32 (ASYNCcnt) to stage halo tiles into LDS
//   * s_wait_asynccnt 0 drain before the workgroup barrier
//   * non-temporal streaming stores for the write-once 64 MiB output

#define TB 16
#define TW (TB + 2)            // 18 (tile row incl. halo)
#define PLANE (TW * TW)        // 324 floats per depth plane
#define DD 8
#define HH 512
#define WW_ 512
#define NT (DD * PLANE)        // 2592 floats staged per block

__launch_bounds__(256)
__global__ void quadinterp3d_kernel(const float* __restrict__ x,
                                    float* __restrict__ out) {
    __shared__ float tile[NT];

    const int tx = threadIdx.x;
    const int ty = threadIdx.y;
    const int tid = ty * TB + tx;
    const int w0 = blockIdx.x * TB;
    const int h0 = blockIdx.y * TB;
    const int bz = blockIdx.z;                 // b*C + c (C == 1)

    const size_t plane = (size_t)HH * WW_;
    const size_t vol   = (size_t)DD * plane;
    const float* xb = x + (size_t)bz * vol;

    // ---- async-stage the 8 x 18 x 18 halo tile (replicate-clamped) ----
    for (int idx = tid; idx < NT; idx += TB * TB) {
        int d  = idx / PLANE;
        int r  = idx - d * PLANE;
        int hh = r / TW;
        int ww = r - hh * TW;
        int gh = h0 + hh - 1; gh = gh < 0 ? 0 : (gh > HH - 1 ? HH - 1 : gh);
        int gw = w0 + ww - 1; gw = gw < 0 ? 0 : (gw > WW_ - 1 ? WW_ - 1 : gw);
        const float* gp = xb + (size_t)d * plane + (size_t)gh * WW_ + gw;
        unsigned lds = (unsigned)(size_t)(&tile[idx]);   // low 32b of generic ptr = LDS offset
        asm volatile("global_load_async_to_lds_b32 %0, %1, off"
                     :: "v"(lds), "v"(gp) : "memory");
    }
    asm volatile("s_wait_asynccnt 0" ::: "memory");
    __syncthreads();

    const int h = h0 + ty;
    const int w = w0 + tx;
    float* outc = out + (size_t)bz * 3 * vol;            // coords base for this (b,c)
    float* outy = out + (size_t)2 * 3 * vol + (size_t)bz * vol;  // after B*C*3 coord vols
    const float* P0 = &tile[(ty + 1) * TW + (tx + 1)];

#pragma unroll
    for (int d = 0; d < DD; ++d) {
        const int dm = d > 0 ? d - 1 : 0;                // replicate clamp in depth
        const int dp = d < DD - 1 ? d + 1 : DD - 1;
        const float* Pc = P0 + d  * PLANE;
        const float* Pm = P0 + dm * PLANE;
        const float* Pp = P0 + dp * PLANE;

        const float c    = Pc[0];
        const float cxm  = Pc[-1],      cxp  = Pc[1];
        const float cym  = Pc[-TW],     cyp  = Pc[TW];
        const float cmm  = Pc[-TW - 1], cpp_ = Pc[TW + 1];
        const float cpm  = Pc[TW - 1],  cmp_ = Pc[-TW + 1];

        const float m0   = Pm[0];
        const float mxm  = Pm[-1],      mxp  = Pm[1];
        const float mym  = Pm[-TW],     myp  = Pm[TW];
        const float mmm  = Pm[-TW - 1], mpp  = Pm[TW + 1];
        const float mpm  = Pm[TW - 1],  mmp  = Pm[-TW + 1];

        const float p0   = Pp[0];
        const float pxm  = Pp[-1],      pxp  = Pp[1];
        const float pym  = Pp[-TW],     pyp  = Pp[TW];
        const float pmm  = Pp[-TW - 1], ppp  = Pp[TW + 1];
        const float ppm  = Pp[TW - 1],  pmp  = Pp[-TW + 1];

        // first-order gradient b = (gx, gy, gs)
        const float gx = 0.5f * (cxp - cxm);
        const float gy = 0.5f * (cyp - cym);
        const float gs = 0.5f * (p0 - m0);

        // Hessian, mixed terms pre-scaled by 1/4 (as in reference)
        const float dxx = cxm + cxp - 2.0f * c;
        const float dyy = cym + cyp - 2.0f * c;
        const float dss = m0 + p0 - 2.0f * c;
        const float dxy = 0.25f * (cmm + cpp_ - cpm - cmp_);
        const float dys = 0.25f * (mym + pyp - pym - myp);
        const float dxs = 0.25f * (mxm + pxp - pxm - mxp);

        // 3x3x3 NMS. Replicate-clamped reads give the same max as the
        // -inf-padded maxpool: every clamped coord is an in-range voxel that
        // is itself inside the 3x3x3 window, so duplicates never change max.
        float mx = fmaxf(c,  fmaxf(cxm, cxp));
        mx = fmaxf(mx, fmaxf(cym, cyp));
        mx = fmaxf(mx, fmaxf(cmm, cpp_));
        mx = fmaxf(mx, fmaxf(cpm, cmp_));
        mx = fmaxf(mx, fmaxf(m0,  p0));
        mx = fmaxf(mx, fmaxf(mxm, mxp));
        mx = fmaxf(mx, fmaxf(mym, myp));
        mx = fmaxf(mx, fmaxf(mmm, mpp));
        mx = fmaxf(mx, fmaxf(mpm, mmp));
        mx = fmaxf(mx, fmaxf(pxm, pxp));
        mx = fmaxf(mx, fmaxf(pym, pyp));
        mx = fmaxf(mx, fmaxf(pmm, ppp));
        mx = fmaxf(mx, fmaxf(ppm, pmp));
        const bool nms = (c == mx);

        // adjugate / Cramer solve of symmetric 3x3
        const float A11 = dyy * dss - dys * dys;
        const float A12 = dxs * dys - dxy * dss;
        const float A13 = dxy * dys - dxs * dyy;
        const float A22 = dxx * dss - dxs * dxs;
        const float A23 = dxy * dxs - dxx * dys;
        const float A33 = dxx * dyy - dxy * dxy;
        const float det = dxx * A11 + dxy * A12 + dxs * A13;
        const bool solved = (det != 0.0f);
        const bool flag = nms && solved;

        const float rdet = 1.0f / det;
        float rx = flag ? -(A11 * gx + A12 * gy + A13 * gs) * rdet : 0.0f;
        float ry = flag ? -(A12 * gx + A22 * gy + A23 * gs) * rdet : 0.0f;
        float rs = flag ? -(A13 * gx + A23 * gy + A33 * gs) * rdet : 0.0f;
        const float amax = fmaxf(fabsf(rx), fmaxf(fabsf(ry), fabsf(rs)));
        if (amax > 0.7f) { rx = 0.0f; ry = 0.0f; rs = 0.0f; }

        const float yv = c + 0.5f * (gx * rx + gy * ry + gs * rs)
                       + (flag ? 10.0f : 0.0f);

        const size_t sp = (size_t)d * plane + (size_t)h * WW_ + w;
        // kornia channel quirk: ch0 = d + s_ref, ch1 = w + y_ref, ch2 = h + x_ref
        __builtin_nontemporal_store((float)d + rs, outc + sp);
        __builtin_nontemporal_store((float)w + ry, outc + vol + sp);
        __builtin_nontemporal_store((float)h + rx, outc + 2 * vol + sp);
        __builtin_nontemporal_store(yv, outy + sp);
    }
}

extern "C" void kernel_launch(void* const* d_in, const int* in_sizes, int n_in,
                              void* d_out, int out_size, void* d_ws, size_t ws_size,
                              hipStream_t stream) {
    const float* x = (const float*)d_in[0];
    float* out = (float*)d_out;
    (void)in_sizes; (void)n_in; (void)out_size; (void)d_ws; (void)ws_size;

    dim3 block(TB, TB, 1);
    dim3 grid(WW_ / TB, HH / TB, 2 /* B*C */);
    quadinterp3d_kernel<<<grid, block, 0, stream>>>(x, out);
}